// ModuleDepthFlowProj_773094113864
// MI455X (gfx1250) — compile-verified
//
#include <hip/hip_runtime.h>
#include <cstdint>

// Problem constants (match the reference).
constexpr int kB = 8;
constexpr int kH = 1080;
constexpr int kW = 1920;
constexpr int kHW = kH * kW;            // 2,073,600
constexpr int kN  = kB * kHW;           // 16,588,800 (divisible by 256 and 4)

// Native FP32 global atomic add (gfx1250 GLOBAL_ATOMIC_ADD_F32 is RNE,
// denorm-preserving -> numerically exact for this use). unsafeAtomicAdd is
// the HIP spelling that permits the hardware instruction on device-local
// memory (our buffers are plain hipMalloc device memory).
__device__ __forceinline__ void atomAddF32(float* p, float v) {
  unsafeAtomicAdd(p, v);
}

// ---------------------------------------------------------------------------
// Pass 1: depth-weighted forward splat with FP32 global atomics.
// outAcc = d_out reused as the (B,2,H,W) accumulator; accW = weight buffer.
// (Defined first so the disasm snippet shows the atomic lowering.)
__global__ void splat_kernel(const float* __restrict__ flow,
                             const float* __restrict__ depth,
                             float* __restrict__ outAcc,
                             float* __restrict__ accW) {
  int i = blockIdx.x * blockDim.x + threadIdx.x;
  if (i >= kN) return;
  int b   = i / kHW;
  int rem = i - b * kHW;
  int y   = rem / kW;
  int x   = rem - y * kW;

  const float* f = flow + (size_t)b * 2 * kHW;
  float fx = f[rem];
  float fy = f[kHW + rem];
  float x2 = (float)x + fx;
  float y2 = (float)y + fy;
  // Invalid pixels contribute exactly 0 in the reference -> skip.
  if (!(x2 >= 0.f && x2 <= (float)(kW - 1) &&
        y2 >= 0.f && y2 <= (float)(kH - 1))) return;

  float w = depth[(size_t)b * kHW + rem];
  int ixL = (int)floorf(x2);  ixL = ixL < 0 ? 0 : (ixL > kW - 1 ? kW - 1 : ixL);
  int iyT = (int)floorf(y2);  iyT = iyT < 0 ? 0 : (iyT > kH - 1 ? kH - 1 : iyT);
  int ixR = ixL + 1 < kW ? ixL + 1 : kW - 1;
  int iyB = iyT + 1 < kH ? iyT + 1 : kH - 1;

  float vx = -fx * w;
  float vy = -fy * w;

  float* ox = outAcc + (size_t)b * 2 * kHW;   // channel 0
  float* oy = ox + kHW;                       // channel 1
  float* cw = accW + (size_t)b * kHW;

  int c0 = iyT * kW + ixL;
  int c1 = iyT * kW + ixR;
  int c2 = iyB * kW + ixL;
  int c3 = iyB * kW + ixR;

  atomAddF32(&ox[c0], vx); atomAddF32(&oy[c0], vy); atomAddF32(&cw[c0], w);
  atomAddF32(&ox[c1], vx); atomAddF32(&oy[c1], vy); atomAddF32(&cw[c1], w);
  atomAddF32(&ox[c2], vx); atomAddF32(&oy[c2], vy); atomAddF32(&cw[c2], w);
  atomAddF32(&ox[c3], vx); atomAddF32(&oy[c3], vy); atomAddF32(&cw[c3], w);
}

// ---------------------------------------------------------------------------
// Zero-fill (vectorized b128 stores).
__global__ void zero_f4(float4* __restrict__ p, int n4) {
  int i = blockIdx.x * blockDim.x + threadIdx.x;
  if (i < n4) p[i] = make_float4(0.f, 0.f, 0.f, 0.f);
}

// ---------------------------------------------------------------------------
// Pass 2: out = acc / max(count, 1), in place on d_out.
// CDNA5 path: stage the weight tile into LDS with the async global->LDS DMA
// (ASYNCcnt-tracked, bypasses VGPR staging), then consume from LDS.
// Grid is exact (kN % 256 == 0) so EXEC is all-ones for the async op.
__global__ void __launch_bounds__(256)
finalize_kernel(float* __restrict__ out, const float* __restrict__ accW) {
  __shared__ float sW[256];
  int i = blockIdx.x * 256 + threadIdx.x;

  // Per-lane async copy: accW[i] -> sW[tid]  (global_load_async_to_lds_b32)
  uint32_t lds = (uint32_t)(uintptr_t)&sW[threadIdx.x];   // low 32 bits = LDS addr
  uint64_t ga  = (uint64_t)(uintptr_t)(accW + i);
  asm volatile("global_load_async_to_lds_b32 %0, %1, off"
               :: "v"(lds), "v"(ga) : "memory");
  asm volatile("s_wait_asynccnt 0" ::: "memory");

  float w = sW[threadIdx.x];
  float denom = w > 0.f ? w : 1.f;

  int b   = i / kHW;
  int rem = i - b * kHW;
  float* ox = out + (size_t)b * 2 * kHW;
  ox[rem]       = ox[rem]       / denom;  // channel 0 (exact div, matches ref)
  ox[kHW + rem] = ox[kHW + rem] / denom;  // channel 1
}

// ---------------------------------------------------------------------------
// Pass 3: hole fill. For count==0 pixels, average nearest filled pixel found
// in each of the 4 axis directions. Filled pixels already hold final values
// in d_out and are never rewritten -> gathers are race-free.
__global__ void fillhole_kernel(const float* __restrict__ accW,
                                float* __restrict__ out) {
  int i = blockIdx.x * blockDim.x + threadIdx.x;
  if (i >= kN) return;
  int b   = i / kHW;
  int rem = i - b * kHW;
  int y   = rem / kW;
  int x   = rem - y * kW;

  const float* cw = accW + (size_t)b * kHW;
  if (cw[rem] > 0.f) return;            // filled: nothing to do

  float* ox = out + (size_t)b * 2 * kHW;
  float* oy = ox + kHW;

  float sx = 0.f, sy = 0.f, s = 0.f;
  const int row = y * kW;

  // left
  { int xi = x - 1;
    while (xi >= 0 && !(cw[row + xi] > 0.f)) --xi;
    if (xi >= 0) { s += 1.f; sx += ox[row + xi]; sy += oy[row + xi]; } }
  // right
  { int xi = x + 1;
    while (xi < kW && !(cw[row + xi] > 0.f)) ++xi;
    if (xi < kW) { s += 1.f; sx += ox[row + xi]; sy += oy[row + xi]; } }
  // up
  { int yi = y - 1;
    while (yi >= 0 && !(cw[yi * kW + x] > 0.f)) --yi;
    if (yi >= 0) { s += 1.f; sx += ox[yi * kW + x]; sy += oy[yi * kW + x]; } }
  // down
  { int yi = y + 1;
    while (yi < kH && !(cw[yi * kW + x] > 0.f)) ++yi;
    if (yi < kH) { s += 1.f; sx += ox[yi * kW + x]; sy += oy[yi * kW + x]; } }

  if (s > 0.f) {
    ox[rem] = sx / s;   // exact division (s can be 3 -> 1/s not exact)
    oy[rem] = sy / s;
  }
  // else: keep 0 (acc was 0, denom was 1 -> out already 0), matches reference.
}

// ---------------------------------------------------------------------------
extern "C" void kernel_launch(void* const* d_in, const int* in_sizes, int n_in,
                              void* d_out, int out_size, void* d_ws, size_t ws_size,
                              hipStream_t stream) {
  const float* flow  = (const float*)d_in[0];   // (B,2,H,W)
  const float* depth = (const float*)d_in[1];   // (B,1,H,W)
  float* out  = (float*)d_out;                  // (B,2,H,W)
  float* accW = (float*)d_ws;                   // kN floats of scratch

  const int n4_out = (2 * kN) / 4;
  const int n4_w   = kN / 4;
  zero_f4<<<(n4_out + 255) / 256, 256, 0, stream>>>((float4*)out, n4_out);
  zero_f4<<<(n4_w   + 255) / 256, 256, 0, stream>>>((float4*)accW, n4_w);

  const int blocks = kN / 256;  // exact: 64800
  splat_kernel   <<<blocks, 256, 0, stream>>>(flow, depth, out, accW);
  finalize_kernel<<<blocks, 256, 0, stream>>>(out, accW);
  fillhole_kernel<<<blocks, 256, 0, stream>>>(accW, out);
}